// BiAttention_81449759801705
// MI455X (gfx1250) — compile-verified
//
#include <hip/hip_runtime.h>
#include <hip/hip_bf16.h>

#define BB 16
#define TT 2048
#define JJ 128
#define DD 1024
#define ST 36   // LDS row stride in floats (multiple of 4 -> 16B-aligned b128 stores)

typedef float v2f __attribute__((ext_vector_type(2)));
typedef float v8f __attribute__((ext_vector_type(8)));
typedef float f4  __attribute__((ext_vector_type(4)));
typedef int  v4i  __attribute__((ext_vector_type(4)));

// ---------------- CDNA5 async global->LDS path (guarded; sync fallback) ----
#if defined(__has_builtin)
#if __has_builtin(__builtin_amdgcn_global_load_async_to_lds_b128)
#define USE_ASYNC 1
#endif
#endif

#ifdef USE_ASYNC
#if __has_builtin(__builtin_amdgcn_s_wait_asynccnt)
#define WAIT_ASYNC() __builtin_amdgcn_s_wait_asynccnt(0)
#else
#define WAIT_ASYNC() asm volatile("s_wait_asynccnt 0x0" ::: "memory")
#endif
#else
#define WAIT_ASYNC() do {} while (0)
#endif

typedef __attribute__((address_space(1))) v4i* gp128;
typedef __attribute__((address_space(3))) v4i* lp128;

__device__ __forceinline__ void cp16(const float* g, float* s) {
#ifdef USE_ASYNC
  __builtin_amdgcn_global_load_async_to_lds_b128((gp128)g, (lp128)s, 0, 0);
#else
  *(f4*)s = *(const f4*)g;
#endif
}

__device__ __forceinline__ v8f wmma4(v2f a, v2f b, v8f c) {
  // D = A(16x4 f32) * B(4x16 f32) + C(16x16 f32)
  return __builtin_amdgcn_wmma_f32_16x16x4_f32(false, a, false, b, (short)0, c,
                                               false, false);
}

// ------------------------------------------------------------- prep: Vaug
// Vaug[b][col][k]: col<128 -> U[b,col,k]*w3[k]; col==128 -> w1[k]; else 0
__global__ __launch_bounds__(256) void k_prep(const float* __restrict__ U,
                                              const float* __restrict__ w,
                                              float* __restrict__ Vaug) {
  long idx = (long)blockIdx.x * 256 + threadIdx.x;   // f4 index
  long e = idx * 4;
  int k = (int)(e % DD);
  long r = e / DD;
  int col = (int)(r % 144);
  int b = (int)(r / 144);
  f4 v;
  if (col < JJ) {
    f4 uq  = *(const f4*)(U + ((long)b * JJ + col) * DD + k);
    f4 w3q = *(const f4*)(w + 2 * DD + k);
    v = uq * w3q;
  } else if (col == JJ) {
    v = *(const f4*)(w + k);
  } else {
    v[0] = v[1] = v[2] = v[3] = 0.f;
  }
  *(f4*)(Vaug + e) = v;
}

// ------------------------------------------------------------- prep: Ut
// Ut[b][d][j] = U[b][j][d]
__global__ __launch_bounds__(256) void k_ut(const float* __restrict__ U,
                                            float* __restrict__ Ut) {
  long idx = (long)blockIdx.x * 256 + threadIdx.x;   // over B * 32(jq) * D
  int d = (int)(idx % DD);
  long g = idx / DD;
  int jq = (int)(g % 32);
  int b = (int)(g / 32);
  f4 v;
  #pragma unroll
  for (int q = 0; q < 4; ++q)
    v[q] = U[((long)b * JJ + jq * 4 + q) * DD + d];
  *(f4*)(Ut + ((long)b * DD + d) * JJ + jq * 4) = v;
}

// ---------------------------------------------------------------- uj = U @ w2
__global__ __launch_bounds__(256) void k_uj(const float* __restrict__ U,
                                            const float* __restrict__ w,
                                            float* __restrict__ uj) {
  int wave = blockIdx.x * 8 + (threadIdx.x >> 5);
  int lane = threadIdx.x & 31;
  const float* u  = U + (long)wave * DD;
  const float* w2 = w + DD;
  float acc = 0.f;
  #pragma unroll 4
  for (int k = lane; k < DD; k += 32) acc += u[k] * w2[k];
  #pragma unroll
  for (int o = 16; o; o >>= 1) acc += __shfl_xor(acc, o, 32);
  if (lane == 0) uj[wave] = acc;
}

// ------------------------- scores GEMM (+hw1 via augmented col) + row softmax
__global__ __launch_bounds__(256) void k_scores(
    const float* __restrict__ H, const float* __restrict__ Vaug,
    const float* __restrict__ wb, const float* __restrict__ uj,
    float* __restrict__ att1, float* __restrict__ mrow) {
  __shared__ float sH[2][128 * ST];   // H tile   [row][k], double-buffered
  __shared__ float sV[2][144 * ST];   // Vaug tile [col][k]
  __shared__ float suj[JJ];

  const int tid = threadIdx.x;
  const int b   = blockIdx.y;
  const int tb  = blockIdx.x;        // 128-row t block
  const int wv  = tid >> 5;
  const int lane = tid & 31;
  const int ln  = lane & 15;
  const int lh  = lane >> 4;

  if (tid < JJ) suj[tid] = uj[b * JJ + tid];

  v8f acc[9];
  #pragma unroll
  for (int q = 0; q < 9; ++q)
    #pragma unroll
    for (int e = 0; e < 8; ++e) acc[q][e] = 0.f;

  const float* Hb = H + ((long)b * TT + (long)tb * 128) * DD;
  const float* Vb = Vaug + (long)b * 144 * DD;

  // stage one k-chunk (32 wide) into buffer `buf`
  auto stage = [&](int k0, int buf) {
    #pragma unroll
    for (int i = 0; i < 4; ++i) {          // H tile: 1024 f4
      int e = tid + i * 256;
      int row = e >> 3, kq = (e & 7) << 2;
      cp16(Hb + (long)row * DD + k0 + kq, &sH[buf][row * ST + kq]);
    }
    #pragma unroll
    for (int i = 0; i < 5; ++i) {          // V tile: 1152 f4
      int e = tid + i * 256;
      if (e < 1152) {
        int row = e >> 3, kq = (e & 7) << 2;
        cp16(Vb + (long)row * DD + k0 + kq, &sV[buf][row * ST + kq]);
      }
    }
  };

  stage(0, 0);
  for (int it = 0; it < DD / 32; ++it) {
    const int cur = it & 1;
    WAIT_ASYNC();
    __syncthreads();
    if (it + 1 < DD / 32) stage((it + 1) * 32, cur ^ 1);
    #pragma unroll
    for (int kk = 0; kk < 32; kk += 4) {
      v2f a = *(const v2f*)(&sH[cur][(wv * 16 + ln) * ST + kk + 2 * lh]);
      #pragma unroll
      for (int ct = 0; ct < 9; ++ct) {
        v2f bf = *(const v2f*)(&sV[cur][(ct * 16 + ln) * ST + kk + 2 * lh]);
        acc[ct] = wmma4(a, bf, acc[ct]);
      }
    }
  }

  const float bias = wb[0];
  #pragma unroll
  for (int i = 0; i < 8; ++i) {
    // hw1 for this lane's row lives at N==128 -> tile 8, lane 0/16, vgpr i
    float hw1 = __shfl(acc[8][i], lane & 16, 32);
    int t = tb * 128 + wv * 16 + i + (lh << 3);
    float s[8];
    float mx = -3.4e38f;
    #pragma unroll
    for (int ct = 0; ct < 8; ++ct) {
      s[ct] = acc[ct][i] + hw1 + suj[ct * 16 + ln] + bias;
      mx = fmaxf(mx, s[ct]);
    }
    #pragma unroll
    for (int o = 1; o < 16; o <<= 1) mx = fmaxf(mx, __shfl_xor(mx, o, 32));
    if (ln == 0) mrow[(long)b * TT + t] = mx;
    float sm = 0.f;
    #pragma unroll
    for (int ct = 0; ct < 8; ++ct) { s[ct] = __expf(s[ct] - mx); sm += s[ct]; }
    #pragma unroll
    for (int o = 1; o < 16; o <<= 1) sm += __shfl_xor(sm, o, 32);
    float r = 1.f / sm;
    long ob = ((long)b * TT + t) * JJ;
    #pragma unroll
    for (int ct = 0; ct < 8; ++ct) att1[ob + ct * 16 + ln] = s[ct] * r;
  }
}

// ----------------------------------------------- att2 = softmax_t(max_j scores)
__global__ __launch_bounds__(256) void k_att2(const float* __restrict__ mrow,
                                              float* __restrict__ att2) {
  __shared__ float red[8];
  __shared__ float bval;
  int b = blockIdx.x, tid = threadIdx.x;
  int lane = tid & 31, wid = tid >> 5;
  const float* mb = mrow + (long)b * TT;
  float v[8];
  float mx = -3.4e38f;
  #pragma unroll
  for (int q = 0; q < 8; ++q) { v[q] = mb[tid + q * 256]; mx = fmaxf(mx, v[q]); }
  #pragma unroll
  for (int o = 16; o; o >>= 1) mx = fmaxf(mx, __shfl_xor(mx, o, 32));
  if (lane == 0) red[wid] = mx;
  __syncthreads();
  if (tid == 0) {
    float r = red[0];
    for (int i = 1; i < 8; ++i) r = fmaxf(r, red[i]);
    bval = r;
  }
  __syncthreads();
  mx = bval;
  float sm = 0.f;
  #pragma unroll
  for (int q = 0; q < 8; ++q) { v[q] = __expf(v[q] - mx); sm += v[q]; }
  #pragma unroll
  for (int o = 16; o; o >>= 1) sm += __shfl_xor(sm, o, 32);
  __syncthreads();
  if (lane == 0) red[wid] = sm;
  __syncthreads();
  if (tid == 0) {
    float r = 0.f;
    for (int i = 0; i < 8; ++i) r += red[i];
    bval = 1.f / r;
  }
  __syncthreads();
  float r = bval;
  #pragma unroll
  for (int q = 0; q < 8; ++q) att2[(long)b * TT + tid + q * 256] = v[q] * r;
}

// -------------------------------------------------- agg2[b,d] = sum_t att2*H
__global__ __launch_bounds__(256) void k_agg2(const float* __restrict__ H,
                                              const float* __restrict__ att2,
                                              float* __restrict__ agg2) {
  __shared__ float sa[TT];
  int b = blockIdx.y;
  int d = blockIdx.x * 256 + threadIdx.x;
  #pragma unroll
  for (int q = 0; q < 8; ++q)
    sa[threadIdx.x + q * 256] = att2[(long)b * TT + threadIdx.x + q * 256];
  __syncthreads();
  const float* Hb = H + (long)b * TT * DD + d;
  float acc = 0.f;
  #pragma unroll 4
  for (int t = 0; t < TT; ++t) acc += sa[t] * Hb[(long)t * DD];
  agg2[b * DD + d] = acc;
}

// ------------------------------- agg1 = att1 @ U (WMMA) fused with G assembly
__global__ __launch_bounds__(256) void k_G(const float* __restrict__ H,
                                           const float* __restrict__ Ut,
                                           const float* __restrict__ att1,
                                           const float* __restrict__ agg2,
                                           float* __restrict__ G) {
  __shared__ float sA[2][128 * ST];   // att1 tile [t][j-chunk]
  __shared__ float sU[2][128 * ST];   // Ut tile   [d-col][j-chunk]
  int tid = threadIdx.x;
  int b = blockIdx.y, tb = blockIdx.x;
  int wv = tid >> 5, lane = tid & 31, ln = lane & 15, lh = lane >> 4;
  const float* Ab = att1 + ((long)b * TT + (long)tb * 128) * JJ;
  const float* Ub = Ut + (long)b * DD * JJ;

  // step s in [0,32): nc = s>>2 (column chunk of 128), k0 = (s&3)*32
  auto stage = [&](int s, int buf) {
    int c0 = (s >> 2) * 128;
    int k0 = (s & 3) * 32;
    #pragma unroll
    for (int i = 0; i < 4; ++i) {          // att1 tile: 1024 f4
      int e = tid + i * 256;
      int row = e >> 3, kq = (e & 7) << 2;
      cp16(Ab + (long)row * JJ + k0 + kq, &sA[buf][row * ST + kq]);
    }
    #pragma unroll
    for (int i = 0; i < 4; ++i) {          // Ut tile: 1024 f4
      int e = tid + i * 256;
      int row = e >> 3, kq = (e & 7) << 2;
      cp16(Ub + (long)(c0 + row) * JJ + k0 + kq, &sU[buf][row * ST + kq]);
    }
  };

  v8f acc[8];
  stage(0, 0);
  for (int s = 0; s < 32; ++s) {
    const int cur = s & 1;
    WAIT_ASYNC();
    __syncthreads();
    if (s + 1 < 32) stage(s + 1, cur ^ 1);
    if ((s & 3) == 0) {
      #pragma unroll
      for (int q = 0; q < 8; ++q)
        #pragma unroll
        for (int e = 0; e < 8; ++e) acc[q][e] = 0.f;
    }
    #pragma unroll
    for (int kk = 0; kk < 32; kk += 4) {
      v2f a = *(const v2f*)(&sA[cur][(wv * 16 + ln) * ST + kk + 2 * lh]);
      #pragma unroll
      for (int ct = 0; ct < 8; ++ct) {
        v2f bf = *(const v2f*)(&sU[cur][(ct * 16 + ln) * ST + kk + 2 * lh]);
        acc[ct] = wmma4(a, bf, acc[ct]);
      }
    }
    if ((s & 3) == 3) {
      // fused epilogue: G = [H, agg1, H*agg1, H*agg2] for this column chunk
      int c0 = (s >> 2) * 128;
      #pragma unroll
      for (int ct = 0; ct < 8; ++ct) {
        int col = c0 + ct * 16 + ln;
        float a2 = agg2[b * DD + col];
        #pragma unroll
        for (int i = 0; i < 8; ++i) {
          int t = tb * 128 + wv * 16 + i + (lh << 3);
          float h  = H[((long)b * TT + t) * DD + col];
          float a1 = acc[ct][i];
          long go = ((long)b * TT + t) * (4 * DD);
          G[go + col]          = h;
          G[go + DD + col]     = a1;
          G[go + 2 * DD + col] = h * a1;
          G[go + 3 * DD + col] = h * a2;
        }
      }
    }
  }
}

extern "C" void kernel_launch(void* const* d_in, const int* in_sizes, int n_in,
                              void* d_out, int out_size, void* d_ws, size_t ws_size,
                              hipStream_t stream) {
  const float* H  = (const float*)d_in[0];
  const float* U  = (const float*)d_in[1];
  const float* w  = (const float*)d_in[2];
  const float* wb = (const float*)d_in[3];

  float* out  = (float*)d_out;
  float* att1 = out;                              // B*T*J
  float* att2 = out + (long)BB * TT * JJ;         // B*T
  float* G    = att2 + (long)BB * TT;             // B*T*4D

  float* ws    = (float*)d_ws;
  float* ws_m  = ws;                              // B*T        (row max)
  float* ws_uj = ws_m + (long)BB * TT;            // B*J        (U @ w2)
  float* ws_a2 = ws_uj + (long)BB * JJ;           // B*D        (agg2)
  float* ws_V  = ws_a2 + (long)BB * DD;           // B*144*D    (Vaug)
  float* ws_Ut = ws_V + (long)BB * 144 * DD;      // B*D*J      (U transposed)

  k_prep  <<<dim3(BB * 144 * DD / 4 / 256), dim3(256), 0, stream>>>(U, w, ws_V);
  k_ut    <<<dim3(BB * 32 * DD / 256),      dim3(256), 0, stream>>>(U, ws_Ut);
  k_uj    <<<dim3(BB * JJ / 8),             dim3(256), 0, stream>>>(U, w, ws_uj);
  k_scores<<<dim3(TT / 128, BB),   dim3(256), 0, stream>>>(H, ws_V, wb, ws_uj, att1, ws_m);
  k_att2  <<<dim3(BB),             dim3(256), 0, stream>>>(ws_m, att2);
  k_agg2  <<<dim3(DD / 256, BB),   dim3(256), 0, stream>>>(H, att2, ws_a2);
  k_G     <<<dim3(TT / 128, BB),   dim3(256), 0, stream>>>(H, ws_Ut, att1, ws_a2, G);
}